// DSNAS_38560216383905
// MI455X (gfx1250) — compile-verified
//
#include <hip/hip_runtime.h>
#include <hip/hip_bf16.h>

// DSNAS interaction-search op for MI455X (gfx1250), wave32 + WMMA f32 16x16x4.
//
// C=22 features, V=12 vocab, D=64, B=2048, P=5 softmax-weighted ops, out (B,2).
//
// Decomposition:
//   add + concat + (max+min)/2   -> linear term  sum_{c,d} e[c,b,d] * A_lin[c,d,o]
//   mul                          -> bilinear GEMM per (d,o): M'[i,b] = W1[i,:,d,o] @ e[:,b,d]
//                                   (V_WMMA_F32_16X16X4_F32, M=i pad 32, N=16 batches, K=j pad 24)
//                                   then stage-B: out += sum_i e[i,b,d]*M'[i,b]
//   (max-min)/2 -> |e_i-e_j|     -> irreducibly pairwise; unrolled 231-pair VALU with
//                                   pre-symmetrized weights.
//
// Software pipelining per d-iteration: issue 6x b128 e-loads + 24x b64 A-fragment
// loads first, hide their latency under the ~700-op abs/linear VALU block, then run
// the 24 WMMAs on resolved registers.

#define CC    22
#define VV    12
#define DD    64
#define BB    2048
#define NPAIR 231   // C*(C-1)/2
#define DCHUNK 4    // d's per wave; grid.y = DD/DCHUNK

typedef float v2f __attribute__((ext_vector_type(2)));
typedef float v4f __attribute__((ext_vector_type(4)));
typedef float v8f __attribute__((ext_vector_type(8)));

// ---------- helpers ----------
__device__ __forceinline__ void softmax5(const float* __restrict__ la, float w[5]) {
    float m = la[0];
    #pragma unroll
    for (int k = 1; k < 5; ++k) m = fmaxf(m, la[k]);
    float s = 0.f, e[5];
    #pragma unroll
    for (int k = 0; k < 5; ++k) { e[k] = __expf(la[k] - m); s += e[k]; }
    float inv = 1.f / s;
    #pragma unroll
    for (int k = 0; k < 5; ++k) w[k] = e[k] * inv;
}

// ---------- kernel: zero the output (out is poisoned by the harness) ----------
__global__ void dsnas_zero_out(float* __restrict__ out) {
    out[blockIdx.x * 256 + threadIdx.x] = 0.f;
}

// ---------- kernel: gather embeddings Et[d][b][c:32] (row per (d,b), 128B rows) ----------
__global__ void dsnas_gather(const float* __restrict__ emb,
                             const int*   __restrict__ idx,
                             float*       __restrict__ Et) {
    const int b = blockIdx.x * 256 + threadIdx.x;   // 0..2047
    const int d = blockIdx.y;                       // 0..63
    float v[24];
    #pragma unroll
    for (int c = 0; c < CC; ++c) {
        const int r = idx[c * BB + b];
        v[c] = emb[(c * VV + r) * DD + d];
    }
    v[22] = 0.f; v[23] = 0.f;
    v4f* o = (v4f*)(Et + ((size_t)d * BB + b) * 32);
    #pragma unroll
    for (int k = 0; k < 6; ++k) {
        v4f q = { v[4 * k + 0], v[4 * k + 1], v[4 * k + 2], v[4 * k + 3] };
        o[k] = q;
    }
}

// ---------- kernel: Wmul[o][d][i:32][j:24] = w1 * W_ops[1,i,j,d,o] (zero padded) ----------
__global__ void dsnas_prep_mul(const float* __restrict__ Wops,
                               const float* __restrict__ la,
                               float*       __restrict__ Wmul) {
    const int tid = blockIdx.x * 256 + threadIdx.x;   // < 2*64*32*24 = 98304
    const int j = tid % 24;
    const int i = (tid / 24) % 32;
    const int d = (tid / (24 * 32)) % 64;
    const int o = tid / (24 * 32 * 64);
    float w[5]; softmax5(la, w);
    float v = 0.f;
    if (i < CC && j < CC)
        v = w[1] * Wops[((((size_t)1 * CC + i) * CC + j) * DD + d) * 2 + o];
    Wmul[tid] = v;
}

// ---------- kernel: WabsS[p][d][o] = 0.25*(w2*(W2[i,j]+W2[j,i]) - w3*(W3[i,j]+W3[j,i])) ----------
// (0.5 from max/min folding, 0.5 from half-group duplication in the main kernel)
__global__ void dsnas_prep_abs(const float* __restrict__ Wops,
                               const float* __restrict__ la,
                               float*       __restrict__ Wabs) {
    const int tid = blockIdx.x * 256 + threadIdx.x;
    if (tid >= NPAIR * DD * 2) return;
    const int o = tid % 2;
    const int d = (tid / 2) % DD;
    const int p = tid / (2 * DD);
    int i = 0, rem = p;
    while (rem >= (CC - 1) - i) { rem -= (CC - 1) - i; ++i; }
    const int j = i + 1 + rem;
    float w[5]; softmax5(la, w);
    #define WK(k, a, bb) Wops[((((size_t)(k) * CC + (a)) * CC + (bb)) * DD + d) * 2 + o]
    const float v = 0.25f * ( w[2] * (WK(2, i, j) + WK(2, j, i))
                            - w[3] * (WK(3, i, j) + WK(3, j, i)) );
    #undef WK
    Wabs[tid] = v;
}

// ---------- kernel: Alin[o][d][c:32] folded linear weights (pre-halved, zero padded) ----------
__global__ void dsnas_prep_lin(const float* __restrict__ Wops,
                               const float* __restrict__ Wcat,
                               const float* __restrict__ la,
                               float*       __restrict__ Alin) {
    const int tid = blockIdx.x * 256 + threadIdx.x;   // < 2*64*32 = 4096
    const int c = tid % 32;
    const int d = (tid / 32) % DD;
    const int o = tid / (32 * DD);
    float v = 0.f;
    if (c < CC) {
        float w[5]; softmax5(la, w);
        float rs0 = 0.f, cs0 = 0.f, rs2 = 0.f, cs2 = 0.f, rs3 = 0.f, cs3 = 0.f;
        float rsl = 0.f, csh = 0.f;
        #pragma unroll 1
        for (int x = 0; x < CC; ++x) {
            rs0 += Wops[((((size_t)0 * CC + c) * CC + x) * DD + d) * 2 + o];
            cs0 += Wops[((((size_t)0 * CC + x) * CC + c) * DD + d) * 2 + o];
            rs2 += Wops[((((size_t)2 * CC + c) * CC + x) * DD + d) * 2 + o];
            cs2 += Wops[((((size_t)2 * CC + x) * CC + c) * DD + d) * 2 + o];
            rs3 += Wops[((((size_t)3 * CC + c) * CC + x) * DD + d) * 2 + o];
            cs3 += Wops[((((size_t)3 * CC + x) * CC + c) * DD + d) * 2 + o];
            rsl += Wcat[(((size_t)c * CC + x) * (2 * DD) + d) * 2 + o];
            csh += Wcat[(((size_t)x * CC + c) * (2 * DD) + DD + d) * 2 + o];
        }
        // 0.5 outer factor: half-group duplication in the main kernel
        v = 0.5f * ( w[0] * (rs0 + cs0)
                   + 0.5f * (w[2] * (rs2 + cs2) + w[3] * (rs3 + cs3))
                   + w[4] * (rsl + csh) );
    }
    Alin[tid] = v;
}

// ---------- main kernel: WMMA GEMM (mul term) + abs pair term + linear term ----------
__global__ __launch_bounds__(32)
void dsnas_main(const float* __restrict__ Et,
                const float* __restrict__ Wmul,
                const float* __restrict__ Wabs,
                const float* __restrict__ Alin,
                float*       __restrict__ out) {
    const int lane = threadIdx.x;         // 0..31
    const int m    = lane & 15;           // batch within tile == WMMA N / A-row index
    const int hg   = lane >> 4;           // half-group: selects K (and M) sub-rows
    const int b    = blockIdx.x * 16 + m; // this lane's batch
    const int d0   = blockIdx.y * DCHUNK; // d-chunk

    float pr[2] = {0.f, 0.f};
    const v2f* __restrict__ WmulV = (const v2f*)Wmul;
    const v4f* __restrict__ EtV   = (const v4f*)Et;

    #pragma unroll 1
    for (int d = d0; d < d0 + DCHUNK; ++d) {
        // prefetch next d's embedding row for this lane (speculative; OOB dropped)
        __builtin_prefetch(Et + ((size_t)(d + 1) * BB + b) * 32, 0, 1);

        // ---- issue all loads up front (distinct registers -> latency overlapped) ----
        // e[0..23] = Et[d][b][*] as 6 x b128
        v4f e4[6];
        #pragma unroll
        for (int k = 0; k < 6; ++k) e4[k] = EtV[((size_t)d * BB + b) * 8 + k];

        // A fragments for all (o, t, s): 24 x b64, b-tile-invariant weights from L2
        v2f af[2][2][6];
        #pragma unroll
        for (int o = 0; o < 2; ++o)
            #pragma unroll
            for (int t = 0; t < 2; ++t)
                #pragma unroll
                for (int s = 0; s < 6; ++s)
                    af[o][t][s] =
                        WmulV[((o * DD + d) * 32 + 16 * t + m) * 12 + 2 * s + hg];

        float e[24];
        #pragma unroll
        for (int k = 0; k < 6; ++k) {
            e[4 * k + 0] = e4[k].x; e[4 * k + 1] = e4[k].y;
            e[4 * k + 2] = e4[k].z; e[4 * k + 3] = e4[k].w;
        }

        // --- linear term (duplicated across half-groups; Alin pre-halved) ---
        #pragma unroll
        for (int c = 0; c < CC; ++c) {
            pr[0] = fmaf(e[c], Alin[(0 * DD + d) * 32 + c], pr[0]);
            pr[1] = fmaf(e[c], Alin[(1 * DD + d) * 32 + c], pr[1]);
        }

        // --- |e_i - e_j| pair term (duplicated; Wabs pre-quartered; 231 pairs) ---
        // big VALU block: hides the in-flight A-fragment load latency
        #pragma unroll
        for (int i = 0; i < CC - 1; ++i) {
            #pragma unroll
            for (int j = i + 1; j < CC; ++j) {
                const int p = i * (CC - 1) - (i * (i - 1)) / 2 + (j - i - 1);
                const float t = fabsf(e[i] - e[j]);
                pr[0] = fmaf(t, Wabs[(p * DD + d) * 2 + 0], pr[0]);
                pr[1] = fmaf(t, Wabs[(p * DD + d) * 2 + 1], pr[1]);
            }
        }

        // --- mul term: per (d,o) GEMM  M'[i, b-tile] = Wmul[i,:,d,o] @ e[:, b-tile] ---
        // B operand (K=4 x N=16, f32): VGPR0 = rows {k0 | k0+2}, VGPR1 = rows {k0+1 | k0+3}
        v2f bf[6];
        #pragma unroll
        for (int s = 0; s < 6; ++s) {
            bf[s].x = hg ? e[4 * s + 2] : e[4 * s + 0];
            bf[s].y = hg ? e[4 * s + 3] : e[4 * s + 1];
        }

        #pragma unroll
        for (int o = 0; o < 2; ++o) {
            #pragma unroll
            for (int t = 0; t < 2; ++t) {          // M tiles: i = 0..15, 16..31(pad)
                v8f acc = {0.f, 0.f, 0.f, 0.f, 0.f, 0.f, 0.f, 0.f};
                #pragma unroll
                for (int s = 0; s < 6; ++s) {      // K steps of 4 over j (pad 24)
                    // A operand (M=16 x K=4): lane m holds row i=16t+m,
                    // VGPRs = cols {4s+2hg, 4s+2hg+1}
                    acc = __builtin_amdgcn_wmma_f32_16x16x4_f32(
                        false, af[o][t][s], false, bf[s], (short)0, acc, false, false);
                }
                // stage B: out += sum_i e[i] * M'[i,b]; D rows for this lane are
                // i = 16t + v + 8*hg; halves are complementary (combined via shfl).
                #pragma unroll
                for (int v = 0; v < 8; ++v) {
                    float ei;
                    if (t == 0) ei = hg ? e[v + 8] : e[v];
                    else        ei = hg ? 0.f     : e[16 + v];  // e[22],e[23]==0
                    pr[o] = fmaf(acc[v], ei, pr[o]);
                }
            }
        }
    }

    // combine lane l with l+16 (same b, complementary/duplicated partials)
    pr[0] += __shfl_xor(pr[0], 16, 32);
    pr[1] += __shfl_xor(pr[1], 16, 32);
    if (lane < 16) {
        atomicAdd(&out[(size_t)b * 2 + 0], pr[0]);
        atomicAdd(&out[(size_t)b * 2 + 1], pr[1]);
    }
}

// ---------- launch ----------
extern "C" void kernel_launch(void* const* d_in, const int* in_sizes, int n_in,
                              void* d_out, int out_size, void* d_ws, size_t ws_size,
                              hipStream_t stream) {
    (void)in_sizes; (void)n_in; (void)out_size; (void)ws_size;

    const float* emb  = (const float*)d_in[0];  // (22,12,64)
    const float* Wops = (const float*)d_in[1];  // (4,22,22,64,2)
    const float* Wcat = (const float*)d_in[2];  // (22,22,128,2)
    const float* la   = (const float*)d_in[3];  // (1,5)
    const int*   idx  = (const int*)  d_in[4];  // (22,2048)
    float*       out  = (float*)d_out;          // (2048,2)

    float* ws   = (float*)d_ws;
    float* Et   = ws;                           // 64*2048*32      = 4,194,304 f
    float* Wmul = Et + (size_t)DD * BB * 32;    // 2*64*32*24      =    98,304 f
    float* Wabs = Wmul + 2 * DD * 32 * 24;      // 231*64*2        =    29,568 f
    float* Alin = Wabs + NPAIR * DD * 2;        // 2*64*32         =     4,096 f

    dsnas_zero_out<<<16, 256, 0, stream>>>(out);                       // 4096
    dsnas_gather  <<<dim3(BB / 256, DD), 256, 0, stream>>>(emb, idx, Et);
    dsnas_prep_mul<<<(2 * DD * 32 * 24) / 256, 256, 0, stream>>>(Wops, la, Wmul);
    dsnas_prep_abs<<<(NPAIR * DD * 2 + 255) / 256, 256, 0, stream>>>(Wops, la, Wabs);
    dsnas_prep_lin<<<(2 * DD * 32) / 256, 256, 0, stream>>>(Wops, Wcat, la, Alin);
    dsnas_main    <<<dim3(BB / 16, DD / DCHUNK), 32, 0, stream>>>(Et, Wmul, Wabs, Alin, out);
}